// PillarVFE_24292335026905
// MI455X (gfx1250) — compile-verified
//
#include <hip/hip_runtime.h>
#include <hip/hip_bf16.h>

typedef __attribute__((ext_vector_type(2))) float v2f;
typedef __attribute__((ext_vector_type(8))) float v8f;

#define N_PILLARS 100000
#define PPTS      32
#define F_IN      4
#define C_OUTC    64
#define C_FEATC   10
#define KPAD      12
#define NXC       432
#define NYC       496
#define GC        (432 * 496)       // NZ*NX*NY, NZ=1
#define BN_EPS_C  0.001f
// voxel geometry: x_off = VX/2+X0 = 0.08 ; y_off = VY/2+Y0 = -39.60 ; z_off = VZ/2+Z0 = -1.0
#define VXC 0.16f
#define VYC 0.16f
#define VZC 4.0f
#define XOFF 0.08f
#define YOFF (-39.60f)
#define ZOFF (-1.0f)

// ---------------------------------------------------------------- zero canvas
__global__ void pillar_zero_f32(float* __restrict__ out, size_t n) {
  size_t i = (size_t)blockIdx.x * blockDim.x + threadIdx.x;
  size_t stride = (size_t)gridDim.x * blockDim.x;
  size_t n4 = n >> 2;
  float4* o4 = (float4*)out;
  for (size_t j = i; j < n4; j += stride)
    o4[j] = make_float4(0.f, 0.f, 0.f, 0.f);
  if (i < (n & 3)) out[(n4 << 2) + i] = 0.f;
}

// ---------------------------------------------------------------- main kernel
// one wave (32 lanes) per pillar; 8 waves per block
__launch_bounds__(256)
__global__ void pillar_vfe_wmma(const float* __restrict__ voxels,
                                const int*   __restrict__ coords,
                                const int*   __restrict__ num_points,
                                const float* __restrict__ Wm,
                                const float* __restrict__ gamma,
                                const float* __restrict__ beta,
                                const float* __restrict__ rmean,
                                const float* __restrict__ rvar,
                                float* __restrict__ out) {
  __shared__ __align__(16) float lds_feat[8][PPTS * KPAD];  // 12 KB / block

  const int lane   = threadIdx.x & 31;
  const int wave   = threadIdx.x >> 5;
  const int laneLo = lane & 15;
  const int hi     = lane >> 4;  // 0: lanes 0-15, 1: lanes 16-31

  // ---- B fragments: B[k][n] = W[t*16+n][k], ISA 4x16 f32 layout:
  //   VGPR v, lanes 0-15 -> K = 4c+v ; lanes 16-31 -> K = 4c+v+2
  float Bfrag[4][3][2];
#pragma unroll
  for (int t = 0; t < 4; ++t)
#pragma unroll
    for (int c = 0; c < 3; ++c)
#pragma unroll
      for (int v = 0; v < 2; ++v) {
        int k  = 4 * c + v + 2 * hi;
        int ch = t * 16 + laneLo;
        Bfrag[t][c][v] = (k < C_FEATC) ? Wm[ch * C_FEATC + k] : 0.0f;
      }

  // ---- fused BN params; D-tile channel for this lane = t*16 + (lane&15)
  float sc[4], bi[4];
#pragma unroll
  for (int t = 0; t < 4; ++t) {
    int ch = t * 16 + laneLo;
    float s = gamma[ch] * rsqrtf(rvar[ch] + BN_EPS_C);
    sc[t] = s;
    bi[t] = beta[ch] - rmean[ch] * s;
  }

  float* myfeat = lds_feat[wave];

  for (int pillar = blockIdx.x * 8 + wave; pillar < N_PILLARS;
       pillar += gridDim.x * 8) {
    // ---- per-point load (lane == point)
    const float* vp = voxels + (size_t)pillar * (PPTS * F_IN) + lane * F_IN;
    float v0 = vp[0], v1 = vp[1], v2 = vp[2], v3 = vp[3];
    int np = num_points[pillar];
    int cb = coords[pillar * 4 + 0];
    int cz = coords[pillar * 4 + 1];
    int cy = coords[pillar * 4 + 2];
    int cx = coords[pillar * 4 + 3];

    // ---- mean over ALL 32 points (reference sums unmasked) / num_points
    float sx = v0, sy = v1, sz = v2;
#pragma unroll
    for (int off = 16; off >= 1; off >>= 1) {
      sx += __shfl_xor(sx, off);
      sy += __shfl_xor(sy, off);
      sz += __shfl_xor(sz, off);
    }
    float inv = 1.0f / (float)np;
    float mx = sx * inv, my = sy * inv, mz = sz * inv;

    float fcx = v0 - ((float)cx * VXC + XOFF);
    float fcy = v1 - ((float)cy * VYC + YOFF);
    float fcz = v2 - ((float)cz * VZC + ZOFF);

    float msk = (lane < np) ? 1.0f : 0.0f;
    float f[KPAD];
    f[0] = v0 * msk;  f[1] = v1 * msk;  f[2] = v2 * msk;  f[3] = v3 * msk;
    f[4] = (v0 - mx) * msk;  f[5] = (v1 - my) * msk;  f[6] = (v2 - mz) * msk;
    f[7] = fcx * msk;  f[8] = fcy * msk;  f[9] = fcz * msk;
    f[10] = 0.f;  f[11] = 0.f;

    // ---- stage features row-major [point][feat] in this wave's LDS slab
#pragma unroll
    for (int k = 0; k < KPAD; k += 2)
      *(float2*)&myfeat[lane * KPAD + k] = make_float2(f[k], f[k + 1]);
    asm volatile("s_wait_dscnt 0" ::: "memory");  // same-wave LDS RAW (in-order anyway)

    // ---- A fragments, ISA 16x4 f32 layout: lane m = (lane&15)+16*mt,
    //      VGPR0 = K(4c+2hi), VGPR1 = K(4c+2hi+1)
    v2f A[2][3];
#pragma unroll
    for (int mt = 0; mt < 2; ++mt)
#pragma unroll
      for (int c = 0; c < 3; ++c) {
        int p = mt * 16 + laneLo;
        int k = 4 * c + 2 * hi;
        float2 t2 = *(float2*)&myfeat[p * KPAD + k];
        A[mt][c][0] = t2.x;
        A[mt][c][1] = t2.y;
      }

    // ---- 4 channel tiles x 2 point tiles x 3 K-chunks of WMMA f32 16x16x4
    int idx = cz + cy * NXC + cx;
#pragma unroll
    for (int t = 0; t < 4; ++t) {
      v8f acc0 = {0.f, 0.f, 0.f, 0.f, 0.f, 0.f, 0.f, 0.f};
      v8f acc1 = {0.f, 0.f, 0.f, 0.f, 0.f, 0.f, 0.f, 0.f};
#pragma unroll
      for (int c = 0; c < 3; ++c) {
        v2f Bv;
        Bv[0] = Bfrag[t][c][0];
        Bv[1] = Bfrag[t][c][1];
        acc0 = __builtin_amdgcn_wmma_f32_16x16x4_f32(
            false, A[0][c], false, Bv, (short)0, acc0, false, false);
        acc1 = __builtin_amdgcn_wmma_f32_16x16x4_f32(
            false, A[1][c], false, Bv, (short)0, acc1, false, false);
      }
      // BN + ReLU + max over the 16 rows held by this lane (M=0..7 or 8..15)
      float mval = 0.0f;  // ReLU output >= 0, and all P rows participate
#pragma unroll
      for (int r = 0; r < 8; ++r) {
        mval = fmaxf(mval, fmaxf(acc0[r] * sc[t] + bi[t], 0.f));
        mval = fmaxf(mval, fmaxf(acc1[r] * sc[t] + bi[t], 0.f));
      }
      // merge lane halves: lane n gets max over all 16 M for channel t*16+n
      mval = fmaxf(mval, __shfl_xor(mval, 16));
      if (lane < 16) {
        int ch = t * 16 + lane;
        size_t off = ((size_t)cb * C_OUTC + ch) * (size_t)GC + (size_t)idx;
        out[off] = mval;
      }
    }
  }
}

// ---------------------------------------------------------------- launcher
extern "C" void kernel_launch(void* const* d_in, const int* in_sizes, int n_in,
                              void* d_out, int out_size, void* d_ws, size_t ws_size,
                              hipStream_t stream) {
  const float* voxels     = (const float*)d_in[0];
  const int*   coords     = (const int*)d_in[1];
  const int*   num_points = (const int*)d_in[2];
  const float* Wm         = (const float*)d_in[3];
  const float* gamma      = (const float*)d_in[4];
  const float* beta       = (const float*)d_in[5];
  const float* rmean      = (const float*)d_in[6];
  const float* rvar       = (const float*)d_in[7];
  float* out = (float*)d_out;

  // 1) zero the (B, C_OUT, NY, NX) canvas — dominant traffic, float4 stores
  pillar_zero_f32<<<4096, 256, 0, stream>>>(out, (size_t)out_size);

  // 2) per-pillar VFE + scatter; one wave per pillar, 8 waves per block
  int blocks = (N_PILLARS + 7) / 8;  // 12500
  pillar_vfe_wmma<<<blocks, 256, 0, stream>>>(voxels, coords, num_points, Wm,
                                              gamma, beta, rmean, rvar, out);
}